// GraphProjection_17600775979612
// MI455X (gfx1250) — compile-verified
//
#include <hip/hip_runtime.h>
#include <math.h>

// ---------------------------------------------------------------------------
// GraphProjection for MI455X (gfx1250, wave32).
//   B=4, C=256, H=W=128, BNUM=4 (16 blocks of 32x32), BNOD=8, node_num=128.
// Logits are computed with V_WMMA_F32_16X16X4_F32 using the expansion
//   ||(x-a)/s||^2 = x^2.w1 + x.w2m + c0,  w1=1/s^2, w2m=-2a/s^2, c0=sum a^2/s^2
// ---------------------------------------------------------------------------

typedef __attribute__((ext_vector_type(2))) float v2f;
typedef __attribute__((ext_vector_type(8))) float v8f;

#define DIMC   256
#define HWSZ   16384          // 128*128
#define NPIX   1024           // 32*32 pixels per block
#define GBLK   16             // blocks per image
#define KNOD   8              // nodes per block
#define NODES  128            // total nodes
#define OUT_NODES_ELEMS (4*DIMC*NODES)      // 131072
#define EPS_POOL 1e-9f
#define EPS_NORM 1e-12f

// workspace layout (floats)
#define WS_W1P    0                          // [16][16][256] padded w1
#define WS_W2P    (WS_W1P + GBLK*16*DIMC)    // [16][16][256] padded -2*a*w1
#define WS_C0P    (WS_W2P + GBLK*16*DIMC)    // [16][16]
#define WS_INVS   (WS_C0P + GBLK*16)         // [128][256] 1/sigmoid(sigma)
#define WS_GNODES (WS_INVS + NODES*DIMC)     // [4][128][256] per-node normalized

// ---------------------------------------------------------------------------
// Kernel 1: node parameter prep.  grid = 256 rows (g*16+m), block = 256 (c)
// ---------------------------------------------------------------------------
__global__ void gp_prep_kernel(const float* __restrict__ anchor,
                               const float* __restrict__ sigma,
                               float* __restrict__ ws) {
    __shared__ float red[256];
    const int row = blockIdx.x;       // g*16 + m  (m in 0..15, nodes 8..15 pad)
    const int c   = threadIdx.x;
    const int g   = row >> 4;
    const int m   = row & 15;

    float w1 = 0.f, w2m = 0.f, c0part = 0.f;
    if (m < KNOD) {
        const int kk = g * KNOD + m;
        const float sg  = 1.f / (1.f + expf(-sigma[kk * DIMC + c]));
        const float inv = 1.f / sg;
        const float a   = anchor[kk * DIMC + c];
        w1  = inv * inv;
        w2m = -2.f * a * w1;
        c0part = a * a * w1;
        ws[WS_INVS + kk * DIMC + c] = inv;
    }
    ws[WS_W1P + row * DIMC + c] = w1;
    ws[WS_W2P + row * DIMC + c] = w2m;

    red[c] = c0part;
    __syncthreads();
    for (int s = 128; s > 0; s >>= 1) {
        if (c < s) red[c] += red[c + s];
        __syncthreads();
    }
    if (c == 0) ws[WS_C0P + row] = red[0];
}

// ---------------------------------------------------------------------------
// Kernel 2: per (batch, block) main pass.  grid = 64, block = 256 (8 waves)
// ---------------------------------------------------------------------------
__global__ void __launch_bounds__(256, 1)
gp_main_kernel(const float* __restrict__ x,
               const float* __restrict__ anchor,
               float* __restrict__ ws,
               float* __restrict__ out_sa) {
    __shared__ float lds_assign[NPIX * KNOD];  // 32 KB, [p][k]
    __shared__ float lds_denp[128 * KNOD];     // per (wave,lane<16) partial den
    __shared__ float lds_nodes[KNOD * DIMC];   // 8 KB
    __shared__ float lds_den[KNOD];
    __shared__ float lds_scale[KNOD];

    const int bg   = blockIdx.x;
    const int b    = bg >> 4;
    const int g    = bg & 15;
    const int brow = g >> 2;
    const int bcol = g & 3;
    const int tid  = threadIdx.x;
    const int wave = tid >> 5;
    const int lane = tid & 31;

    const float* xg = x + (size_t)b * DIMC * HWSZ;
    const float* w1row = ws + WS_W1P + (g * 16 + (lane & 15)) * DIMC;
    const float* w2row = ws + WS_W2P + (g * 16 + (lane & 15)) * DIMC;

    // per-lane global spatial offsets for this wave's 8 tiles of 16 pixels
    const int pbase = wave * 128;
    int off[8];
    #pragma unroll
    for (int t = 0; t < 8; ++t) {
        const int pix = pbase + t * 16 + (lane & 15);     // 0..1023 row-major
        const int ph = pix >> 5, pw = pix & 31;
        off[t] = (brow * 32 + ph) * 128 + bcol * 32 + pw;
    }

    // ---------------- phase 1a: logits via f32 WMMA ----------------
    v8f acc[8];
    #pragma unroll
    for (int t = 0; t < 8; ++t) acc[t] = (v8f){0.f,0.f,0.f,0.f,0.f,0.f,0.f,0.f};

    const int hoff = (lane >> 4) << 1;                    // 0 (lo half) / 2 (hi)
    for (int c0 = 0; c0 < DIMC; c0 += 4) {
        const int cA = c0 + hoff;
        const v2f a1 = *(const v2f*)(w1row + cA);         // A-frag: w1
        const v2f a2 = *(const v2f*)(w2row + cA);         // A-frag: -2*a*w1
        const float* xc0 = xg + (size_t)cA * HWSZ;
        const float* xc1 = xc0 + HWSZ;
        #pragma unroll
        for (int t = 0; t < 8; ++t) {
            const float xv0 = xc0[off[t]];
            const float xv1 = xc1[off[t]];
            const v2f bsq = { xv0 * xv0, xv1 * xv1 };
            const v2f bx  = { xv0, xv1 };
            acc[t] = __builtin_amdgcn_wmma_f32_16x16x4_f32(
                        false, a1, false, bsq, (short)0, acc[t], false, false);
            acc[t] = __builtin_amdgcn_wmma_f32_16x16x4_f32(
                        false, a2, false, bx,  (short)0, acc[t], false, false);
        }
    }

    float c0s[KNOD];
    #pragma unroll
    for (int j = 0; j < KNOD; ++j) c0s[j] = ws[WS_C0P + g * 16 + j];

    // ---------------- phase 1b: softmax + assignment ----------------
    float denp[KNOD];
    #pragma unroll
    for (int j = 0; j < KNOD; ++j) denp[j] = 0.f;

    float* sa_b = out_sa + (size_t)b * NODES * NPIX;
    if (lane < 16) {
        #pragma unroll
        for (int t = 0; t < 8; ++t) {
            const int p = pbase + t * 16 + lane;
            float lg[KNOD], mx = -3.4e38f;
            #pragma unroll
            for (int j = 0; j < KNOD; ++j) {
                lg[j] = -0.5f * (acc[t][j] + c0s[j]);
                mx = fmaxf(mx, lg[j]);
            }
            float s = 0.f;
            #pragma unroll
            for (int j = 0; j < KNOD; ++j) { lg[j] = expf(lg[j] - mx); s += lg[j]; }
            const float invsum = 1.f / s;
            #pragma unroll
            for (int j = 0; j < KNOD; ++j) {
                const float a = lg[j] * invsum;
                denp[j] += a;
                lds_assign[p * KNOD + j] = a;
                sa_b[(g * KNOD + j) * NPIX + p] = a;
            }
        }
        #pragma unroll
        for (int j = 0; j < KNOD; ++j)
            lds_denp[(wave * 16 + lane) * KNOD + j] = denp[j];
    }
    __syncthreads();
    if (tid < KNOD) {                 // deterministic den reduction
        float s = 0.f;
        for (int r = 0; r < 128; ++r) s += lds_denp[r * KNOD + tid];
        lds_den[tid] = s;
    }
    __syncthreads();

    // ---------------- phase 2: weighted pooling (thread = channel) ----------
    const int c = tid;
    float den0[KNOD], dinv[KNOD];
    #pragma unroll
    for (int j = 0; j < KNOD; ++j) {
        den0[j] = lds_den[j];
        dinv[j] = 1.f / (den0[j] + EPS_POOL);
    }

    float S1[KNOD];
    #pragma unroll
    for (int j = 0; j < KNOD; ++j) S1[j] = 0.f;

    const float* xc = xg + (size_t)c * HWSZ;
    for (int p = 0; p < NPIX; p += 4) {
        const int ph = p >> 5, pw = p & 31;
        const float4 xq = *(const float4*)(xc + (brow * 32 + ph) * 128 + bcol * 32 + pw);
        const float xv[4] = { xq.x, xq.y, xq.z, xq.w };
        #pragma unroll
        for (int q = 0; q < 4; ++q) {
            const float4 alo = *(const float4*)&lds_assign[(p + q) * KNOD];
            const float4 ahi = *(const float4*)&lds_assign[(p + q) * KNOD + 4];
            S1[0] = fmaf(alo.x, xv[q], S1[0]);
            S1[1] = fmaf(alo.y, xv[q], S1[1]);
            S1[2] = fmaf(alo.z, xv[q], S1[2]);
            S1[3] = fmaf(alo.w, xv[q], S1[3]);
            S1[4] = fmaf(ahi.x, xv[q], S1[4]);
            S1[5] = fmaf(ahi.y, xv[q], S1[5]);
            S1[6] = fmaf(ahi.z, xv[q], S1[6]);
            S1[7] = fmaf(ahi.w, xv[q], S1[7]);
        }
    }

    float nv[KNOD];
    #pragma unroll
    for (int j = 0; j < KNOD; ++j) {
        const int kk = g * KNOD + j;
        const float invs = ws[WS_INVS + kk * DIMC + c];
        const float av   = anchor[kk * DIMC + c];
        nv[j] = invs * (S1[j] - av * den0[j]) * dinv[j];
        lds_nodes[j * DIMC + c] = nv[j];
    }
    __syncthreads();
    if (tid < KNOD) {                 // per-node L2 norm
        float s = 0.f;
        for (int cc = 0; cc < DIMC; ++cc) {
            const float v = lds_nodes[tid * DIMC + cc];
            s = fmaf(v, v, s);
        }
        lds_scale[tid] = 1.f / fmaxf(sqrtf(s), EPS_NORM);
    }
    __syncthreads();
    float* gn = ws + WS_GNODES + (size_t)b * NODES * DIMC;
    #pragma unroll
    for (int j = 0; j < KNOD; ++j)
        gn[(g * KNOD + j) * DIMC + c] = nv[j] * lds_scale[j];
}

// ---------------------------------------------------------------------------
// Kernel 3: global L2 norm per batch + transpose.  grid = 4, block = 256
// ---------------------------------------------------------------------------
__global__ void gp_finalize_kernel(const float* __restrict__ ws,
                                   float* __restrict__ out_nodes) {
    __shared__ float red[256];
    const int b   = blockIdx.x;
    const int tid = threadIdx.x;
    const float* gn = ws + WS_GNODES + (size_t)b * NODES * DIMC;

    float s = 0.f;
    for (int i = tid; i < NODES * DIMC; i += 256) {
        const float v = gn[i];
        s = fmaf(v, v, s);
    }
    red[tid] = s;
    __syncthreads();
    for (int st = 128; st > 0; st >>= 1) {
        if (tid < st) red[tid] += red[tid + st];
        __syncthreads();
    }
    const float scale = 1.f / fmaxf(sqrtf(red[0]), EPS_NORM);

    float* ob = out_nodes + (size_t)b * DIMC * NODES;
    for (int i = tid; i < NODES * DIMC; i += 256) {
        const int kk = i >> 8;        // i = kk*256 + c
        const int c  = i & 255;
        ob[c * NODES + kk] = gn[i] * scale;   // [C][node]
    }
}

// ---------------------------------------------------------------------------
extern "C" void kernel_launch(void* const* d_in, const int* in_sizes, int n_in,
                              void* d_out, int out_size, void* d_ws, size_t ws_size,
                              hipStream_t stream) {
    (void)in_sizes; (void)n_in; (void)out_size; (void)ws_size;
    const float* x      = (const float*)d_in[0];
    const float* anchor = (const float*)d_in[1];
    const float* sigma  = (const float*)d_in[2];
    float* ws        = (float*)d_ws;
    float* out_nodes = (float*)d_out;                    // [4][256][128]
    float* out_sa    = (float*)d_out + OUT_NODES_ELEMS;  // [4][128][1024]

    gp_prep_kernel<<<GBLK * 16, 256, 0, stream>>>(anchor, sigma, ws);
    gp_main_kernel<<<4 * GBLK, 256, 0, stream>>>(x, anchor, ws, out_sa);
    gp_finalize_kernel<<<4, 256, 0, stream>>>(ws, out_nodes);
}